// HeteroGraphGNN_20581483282605
// MI455X (gfx1250) — compile-verified
//
#include <hip/hip_runtime.h>
#include <cstdint>
#include <cstddef>

// ---------------- CDNA5 types ----------------
typedef float v2f __attribute__((ext_vector_type(2)));
typedef float v8f __attribute__((ext_vector_type(8)));

#define H_HEADS 8
#define GAT_C   128
#define GAT_HC  1024
#define NEG_SLOPE 0.2f
#define LDS_STRIDE 132   // 16 rows x up to 128 K floats, padded; 528B row pitch (16B multiple)

// ---------------- utility kernels ----------------
__global__ void fill_f32_kernel(float* __restrict__ p, long n, float v) {
  long i = (long)blockIdx.x * blockDim.x + threadIdx.x;
  if (i < n) p[i] = v;
}

// agg[dst[e]][c] += x[src[e]][c], D = 1<<dshift
__global__ void scatter_add_kernel(const float* __restrict__ x,
                                   const int* __restrict__ src,
                                   const int* __restrict__ dst,
                                   float* __restrict__ agg,
                                   long total, int dshift, int dmask) {
  long i = (long)blockIdx.x * blockDim.x + threadIdx.x;
  if (i >= total) return;
  long e = i >> dshift;
  int  c = (int)(i & dmask);
  long s = src[e], d = dst[e];
  atomicAdd(agg + (d << dshift) + c, x[(s << dshift) + c]);
}

// ---------------- fp32 WMMA GEMM ----------------
// Block = 128 threads = 4 waves; macro-tile = 16(M) x 128(N); each wave owns a
// 16x32 C tile (two v8f accumulators sharing one A fragment per k-step).
// A tile staged memory->LDS via CDNA5 async-LDS loads (ASYNCcnt path).

// Copy a 16 x K fp32 tile (row-major, row pitch K) into padded LDS rows using
// GLOBAL_LOAD_ASYNC_TO_LDS_B128: each lane moves 16B chunks without VGPR staging.
__device__ __forceinline__ void async_fill_tile(const float* __restrict__ A, int K,
                                                int tile_m, float* ldsBase,
                                                int tid, int nthreads) {
  const int chunksPerRow = K >> 2;          // 16B chunks per row
  const int total = 16 * chunksPerRow;
  const unsigned ldsBaseOff = (unsigned)(uintptr_t)ldsBase;  // LDS aperture: low 32 bits = ds offset
  for (int i = tid; i < total; i += nthreads) {
    int r  = i / chunksPerRow;
    int cc = i - r * chunksPerRow;
    unsigned long long g =
        (unsigned long long)(uintptr_t)(A + (long)(tile_m + r) * K + cc * 4);
    unsigned l = ldsBaseOff + (unsigned)(r * (LDS_STRIDE * 4) + cc * 16);
    asm volatile("global_load_async_to_lds_b128 %0, %1, off"
                 :: "v"(l), "v"(g) : "memory");
  }
}

__device__ __forceinline__ void wmma_panel2(const float* __restrict__ ldsA,
                                            const float* __restrict__ W,
                                            int K, int N, int n0,
                                            int lhalf, int l16,
                                            v8f& acc0, v8f& acc1) {
  for (int k = 0; k < K; k += 4) {
    // ISA layout: A(16x4 f32): VGPR0 = K0 (lanes0-15) / K2 (lanes16-31), VGPR1 = K1/K3.
    // B(4x16 f32): VGPR0 = row K0 (lanes0-15) / K2 (lanes16-31), VGPR1 = K1/K3.
    int kk = k + 2 * lhalf;
    v2f a, b0, b1;
    a.x = ldsA[l16 * LDS_STRIDE + kk];
    a.y = ldsA[l16 * LDS_STRIDE + kk + 1];
    const float* wp = W + (long)kk * N + n0 + l16;
    b0.x = wp[0];
    b0.y = wp[N];
    b1.x = wp[16];
    b1.y = wp[N + 16];
    if (k + 4 < K) __builtin_prefetch(W + (long)(k + 4) * N + n0, 0, 1);
    acc0 = __builtin_amdgcn_wmma_f32_16x16x4_f32(false, a, false, b0, (short)0, acc0,
                                                 false, false);
    acc1 = __builtin_amdgcn_wmma_f32_16x16x4_f32(false, a, false, b1, (short)0, acc1,
                                                 false, false);
  }
}

__global__ void __launch_bounds__(128)
gemm_fused_kernel(const float* __restrict__ A1, int K1, const float* __restrict__ W1,
                  const float* __restrict__ A2, int K2, const float* __restrict__ W2,
                  const float* __restrict__ bias, float* __restrict__ C,
                  int M, int N, int do_relu) {
  __shared__ __align__(16) float ldsA1[16 * LDS_STRIDE];
  __shared__ __align__(16) float ldsA2[16 * LDS_STRIDE];
  const int tid         = threadIdx.x;
  const int tile_m      = blockIdx.x * 16;
  const int tile_n_base = blockIdx.y * 128;

  async_fill_tile(A1, K1, tile_m, ldsA1, tid, 128);
  if (K2 > 0) async_fill_tile(A2, K2, tile_m, ldsA2, tid, 128);
  asm volatile("s_wait_asynccnt 0x0" ::: "memory");  // this wave's async fills done
  __syncthreads();                                   // all waves' fills visible

  const int wave  = tid >> 5;
  const int lane  = tid & 31;
  const int lhalf = lane >> 4;
  const int l16   = lane & 15;
  const int n0    = tile_n_base + wave * 32;  // wave owns cols [n0, n0+32)
  if (n0 >= N) return;  // uniform per wave (all N are multiples of 32) -> EXEC all-ones

  v8f acc0 = {0.f, 0.f, 0.f, 0.f, 0.f, 0.f, 0.f, 0.f};
  v8f acc1 = {0.f, 0.f, 0.f, 0.f, 0.f, 0.f, 0.f, 0.f};
  wmma_panel2(ldsA1, W1, K1, N, n0, lhalf, l16, acc0, acc1);
  if (K2 > 0) wmma_panel2(ldsA2, W2, K2, N, n0, lhalf, l16, acc0, acc1);

  const int   col0 = n0 + l16;
  const float bv0  = bias[col0];
  const float bv1  = bias[col0 + 16];
#pragma unroll
  for (int j = 0; j < 8; ++j) {  // C layout: VGPR j -> row j (lanes0-15) / row j+8
    int   row = tile_m + j + 8 * lhalf;
    float v0  = acc0[j] + bv0;
    float v1  = acc1[j] + bv1;
    if (do_relu) {
      v0 = v0 > 0.f ? v0 : 0.f;
      v1 = v1 > 0.f ? v1 : 0.f;
    }
    C[(long)row * N + col0]      = v0;
    C[(long)row * N + col0 + 16] = v1;
  }
}

// ---------------- GATv2 edge kernels ----------------
// score[e,h] = sum_c att[h,c] * leaky_relu(xl[s,h,c] + xr[d,h,c])
// block = 256 = 8 waves; wave h handles head h; lane covers 4 contiguous channels.
__global__ void __launch_bounds__(256)
gat_score_kernel(const float* __restrict__ xl, const float* __restrict__ xr,
                 const float* __restrict__ att,
                 const int* __restrict__ sidx, const int* __restrict__ didx,
                 float* __restrict__ score, int E) {
  int e = blockIdx.x;
  if (e >= E) return;
  int  h    = threadIdx.x >> 5;
  int  lane = threadIdx.x & 31;
  long s = sidx[e], d = didx[e];
  const float4* l4 = (const float4*)(xl + s * GAT_HC + h * GAT_C);
  const float4* r4 = (const float4*)(xr + d * GAT_HC + h * GAT_C);
  const float4* a4 = (const float4*)(att + h * GAT_C);
  float4 lv = l4[lane], rv = r4[lane], av = a4[lane];
  float sum = 0.f, t;
  t = lv.x + rv.x; sum += av.x * (t > 0.f ? t : NEG_SLOPE * t);
  t = lv.y + rv.y; sum += av.y * (t > 0.f ? t : NEG_SLOPE * t);
  t = lv.z + rv.z; sum += av.z * (t > 0.f ? t : NEG_SLOPE * t);
  t = lv.w + rv.w; sum += av.w * (t > 0.f ? t : NEG_SLOPE * t);
  for (int off = 16; off > 0; off >>= 1) sum += __shfl_down(sum, off, 32);
  if (lane == 0) score[(long)e * H_HEADS + h] = sum;
}

// ordered-uint mapping: monotone in float order (incl. -inf), enables atomicMax
__device__ __forceinline__ unsigned f2ord(float f) {
  unsigned u = __float_as_uint(f);
  return (u & 0x80000000u) ? ~u : (u | 0x80000000u);
}
__device__ __forceinline__ float ord2f(unsigned u) {
  return (u & 0x80000000u) ? __uint_as_float(u & 0x7fffffffu) : __uint_as_float(~u);
}

__global__ void seg_max_kernel(const float* __restrict__ score,
                               const int* __restrict__ didx,
                               unsigned* __restrict__ m, long EH) {
  long i = (long)blockIdx.x * blockDim.x + threadIdx.x;
  if (i >= EH) return;
  long e = i >> 3;
  int  h = (int)(i & 7);
  long d = didx[e];
  atomicMax(m + d * H_HEADS + h, f2ord(score[i]));
}

__global__ void exp_sum_kernel(float* __restrict__ score,
                               const int* __restrict__ didx,
                               const unsigned* __restrict__ m,
                               float* __restrict__ denom, long EH) {
  long i = (long)blockIdx.x * blockDim.x + threadIdx.x;
  if (i >= EH) return;
  long e = i >> 3;
  int  h = (int)(i & 7);
  long d = didx[e];
  float a = __expf(score[i] - ord2f(m[d * H_HEADS + h]));
  score[i] = a;  // keep numerator in place
  atomicAdd(denom + d * H_HEADS + h, a);
}

// out[d,c] += (1/8) * sum_h alpha[e,h] * xl[s,h,c]   (head-mean folded in)
__global__ void __launch_bounds__(128)
gat_aggregate_kernel(const float* __restrict__ xl,
                     const int* __restrict__ sidx, const int* __restrict__ didx,
                     const float* __restrict__ anum, const float* __restrict__ denom,
                     float* __restrict__ out, int E) {
  int e = blockIdx.x;
  if (e >= E) return;
  __shared__ float alpha[H_HEADS];
  long s = sidx[e], d = didx[e];
  if (threadIdx.x < H_HEADS)
    alpha[threadIdx.x] = anum[(long)e * H_HEADS + threadIdx.x] /
                         denom[d * H_HEADS + threadIdx.x];
  __syncthreads();
  int c = threadIdx.x;
  const float* row = xl + s * GAT_HC;
  float acc = 0.f;
#pragma unroll
  for (int h = 0; h < H_HEADS; ++h) acc += alpha[h] * row[h * GAT_C + c];
  atomicAdd(out + d * GAT_C + c, acc * 0.125f);
}

__global__ void bias_act_kernel(float* __restrict__ x, const float* __restrict__ bias,
                                long total, int cmask, int do_relu) {
  long i = (long)blockIdx.x * blockDim.x + threadIdx.x;
  if (i >= total) return;
  float v = x[i] + bias[i & cmask];
  if (do_relu) v = v > 0.f ? v : 0.f;
  x[i] = v;
}

// out[p] = dot64(xs3[ls[p]], xd3[ld[p]]); one wave per pair, 8 pairs / block
__global__ void __launch_bounds__(256)
classifier_kernel(const float* __restrict__ xs3, const float* __restrict__ xd3,
                  const int* __restrict__ ls, const int* __restrict__ ld,
                  float* __restrict__ out, int L) {
  int wave = threadIdx.x >> 5;
  int lane = threadIdx.x & 31;
  int p    = blockIdx.x * 8 + wave;
  if (p >= L) return;
  long s = ls[p], d = ld[p];
  const float* a = xs3 + s * 64;
  const float* b = xd3 + d * 64;
  float acc = a[lane] * b[lane] + a[lane + 32] * b[lane + 32];
  for (int off = 16; off > 0; off >>= 1) acc += __shfl_down(acc, off, 32);
  if (lane == 0) out[p] = acc;
}

// ---------------- host orchestration ----------------
extern "C" void kernel_launch(void* const* d_in, const int* in_sizes, int n_in,
                              void* d_out, int out_size, void* d_ws, size_t ws_size,
                              hipStream_t stream) {
  (void)in_sizes; (void)n_in; (void)out_size; (void)ws_size;
  const float* x_s = (const float*)d_in[0];
  const float* x_d = (const float*)d_in[1];
  const int* es = (const int*)d_in[2];
  const int* ed = (const int*)d_in[3];
  const int* ls = (const int*)d_in[4];
  const int* ld = (const int*)d_in[5];
  const float* W1_rel_sd  = (const float*)d_in[6];
  const float* b1_sd      = (const float*)d_in[7];
  const float* W1_root_sd = (const float*)d_in[8];
  const float* W1_rel_ds  = (const float*)d_in[9];
  const float* b1_ds      = (const float*)d_in[10];
  const float* W1_root_ds = (const float*)d_in[11];
  const float* Wl2_sd = (const float*)d_in[12], *bl2_sd = (const float*)d_in[13];
  const float* Wr2_sd = (const float*)d_in[14], *br2_sd = (const float*)d_in[15];
  const float* att2_sd = (const float*)d_in[16], *bias2_sd = (const float*)d_in[17];
  const float* Wl2_ds = (const float*)d_in[18], *bl2_ds = (const float*)d_in[19];
  const float* Wr2_ds = (const float*)d_in[20], *br2_ds = (const float*)d_in[21];
  const float* att2_ds = (const float*)d_in[22], *bias2_ds = (const float*)d_in[23];
  const float* W3_rel_sd  = (const float*)d_in[24], *b3_sd = (const float*)d_in[25];
  const float* W3_root_sd = (const float*)d_in[26];
  const float* W3_rel_ds  = (const float*)d_in[27], *b3_ds = (const float*)d_in[28];
  const float* W3_root_ds = (const float*)d_in[29];

  constexpr int NS = 10000, ND = 2000, E = 100000, L = 50000;
  constexpr int H1 = 128, H2 = 128, H3 = 64, DS = 128, DD = 64;

  float* w = (float*)d_ws;
  size_t off = 0;
  auto alloc = [&](size_t n) { float* p = w + off; off += (n + 63) & ~(size_t)63; return p; };
  float* xd1   = alloc((size_t)ND * H1);
  float* xs1   = alloc((size_t)NS * H1);
  float* xd2   = alloc((size_t)ND * H2);
  float* xs2   = alloc((size_t)NS * H2);
  float* xd3   = alloc((size_t)ND * H3);
  float* xs3   = alloc((size_t)NS * H3);
  float* agg   = alloc((size_t)NS * 128);     // scatter scratch (reused per phase)
  float* bufA  = alloc((size_t)NS * 1024);    // GAT xl
  float* bufB  = alloc((size_t)NS * 1024);    // GAT xr
  float* score = alloc((size_t)E * H_HEADS);
  float* m_f   = alloc((size_t)NS * H_HEADS); // ordered-uint max buffer
  float* denom = alloc((size_t)NS * H_HEADS);

  dim3 b256(256);
  dim3 b128(128);
  auto fill = [&](float* p, long n, float v) {
    fill_f32_kernel<<<dim3((unsigned)((n + 255) / 256)), b256, 0, stream>>>(p, n, v);
  };

  // ======== Layer 1: GraphConv s->d ========
  fill(agg, (long)ND * H1, 0.f);
  { long tot = (long)E * DS;
    scatter_add_kernel<<<dim3((unsigned)((tot + 255) / 256)), b256, 0, stream>>>(
        x_s, es, ed, agg, tot, 7, 127); }
  gemm_fused_kernel<<<dim3(ND / 16, 1), b128, 0, stream>>>(
      agg, DS, W1_rel_sd, x_d, DD, W1_root_sd, b1_sd, xd1, ND, H1, 1);
  // ======== Layer 1: GraphConv d->s ========
  fill(agg, (long)NS * DD, 0.f);
  { long tot = (long)E * DD;
    scatter_add_kernel<<<dim3((unsigned)((tot + 255) / 256)), b256, 0, stream>>>(
        x_d, ed, es, agg, tot, 6, 63); }
  gemm_fused_kernel<<<dim3(NS / 16, 1), b128, 0, stream>>>(
      agg, DD, W1_rel_ds, x_s, DS, W1_root_ds, b1_ds, xs1, NS, H1, 1);

  // ======== Layer 2: GATv2 s->d ========
  gemm_fused_kernel<<<dim3(NS / 16, 8), b128, 0, stream>>>(
      xs1, H1, Wl2_sd, nullptr, 0, nullptr, bl2_sd, bufA, NS, GAT_HC, 0);
  gemm_fused_kernel<<<dim3(ND / 16, 8), b128, 0, stream>>>(
      xd1, H1, Wr2_sd, nullptr, 0, nullptr, br2_sd, bufB, ND, GAT_HC, 0);
  gat_score_kernel<<<dim3(E), b256, 0, stream>>>(bufA, bufB, att2_sd, es, ed, score, E);
  fill(m_f, (long)ND * H_HEADS, 0.f);   // 0u == below ord(-inf)
  fill(denom, (long)ND * H_HEADS, 0.f);
  seg_max_kernel<<<dim3((E * 8 + 255) / 256), b256, 0, stream>>>(
      score, ed, (unsigned*)m_f, (long)E * 8);
  exp_sum_kernel<<<dim3((E * 8 + 255) / 256), b256, 0, stream>>>(
      score, ed, (const unsigned*)m_f, denom, (long)E * 8);
  fill(xd2, (long)ND * H2, 0.f);
  gat_aggregate_kernel<<<dim3(E), b128, 0, stream>>>(
      bufA, es, ed, score, denom, xd2, E);
  bias_act_kernel<<<dim3((unsigned)(((long)ND * H2 + 255) / 256)), b256, 0, stream>>>(
      xd2, bias2_sd, (long)ND * H2, 127, 1);

  // ======== Layer 2: GATv2 d->s ========
  gemm_fused_kernel<<<dim3(ND / 16, 8), b128, 0, stream>>>(
      xd1, H1, Wl2_ds, nullptr, 0, nullptr, bl2_ds, bufA, ND, GAT_HC, 0);
  gemm_fused_kernel<<<dim3(NS / 16, 8), b128, 0, stream>>>(
      xs1, H1, Wr2_ds, nullptr, 0, nullptr, br2_ds, bufB, NS, GAT_HC, 0);
  gat_score_kernel<<<dim3(E), b256, 0, stream>>>(bufA, bufB, att2_ds, ed, es, score, E);
  fill(m_f, (long)NS * H_HEADS, 0.f);
  fill(denom, (long)NS * H_HEADS, 0.f);
  seg_max_kernel<<<dim3((E * 8 + 255) / 256), b256, 0, stream>>>(
      score, es, (unsigned*)m_f, (long)E * 8);
  exp_sum_kernel<<<dim3((E * 8 + 255) / 256), b256, 0, stream>>>(
      score, es, (const unsigned*)m_f, denom, (long)E * 8);
  fill(xs2, (long)NS * H2, 0.f);
  gat_aggregate_kernel<<<dim3(E), b128, 0, stream>>>(
      bufA, ed, es, score, denom, xs2, E);
  bias_act_kernel<<<dim3((unsigned)(((long)NS * H2 + 255) / 256)), b256, 0, stream>>>(
      xs2, bias2_ds, (long)NS * H2, 127, 1);

  // ======== Layer 3: GraphConv s->d ========
  fill(agg, (long)ND * H2, 0.f);
  { long tot = (long)E * H2;
    scatter_add_kernel<<<dim3((unsigned)((tot + 255) / 256)), b256, 0, stream>>>(
        xs2, es, ed, agg, tot, 7, 127); }
  gemm_fused_kernel<<<dim3(ND / 16, 1), b128, 0, stream>>>(
      agg, H2, W3_rel_sd, xd2, H2, W3_root_sd, b3_sd, xd3, ND, H3, 0);
  // ======== Layer 3: GraphConv d->s ========
  fill(agg, (long)NS * H2, 0.f);
  { long tot = (long)E * H2;
    scatter_add_kernel<<<dim3((unsigned)((tot + 255) / 256)), b256, 0, stream>>>(
        xd2, ed, es, agg, tot, 7, 127); }
  gemm_fused_kernel<<<dim3(NS / 16, 1), b128, 0, stream>>>(
      agg, H2, W3_rel_ds, xs2, H2, W3_root_ds, b3_ds, xs3, NS, H3, 0);

  // ======== Classifier ========
  classifier_kernel<<<dim3((L + 7) / 8), b256, 0, stream>>>(
      xs3, xd3, ls, ld, (float*)d_out, L);
}